// SOMLayer_61134564491922
// MI455X (gfx1250) — compile-verified
//
#include <hip/hip_runtime.h>

typedef __attribute__((ext_vector_type(2))) float v2f;
typedef __attribute__((ext_vector_type(8))) float v8f;

#define T_SEQ   3500
#define L_DIM   256
#define M_DIM   1024
#define ROWS    32          // rows per workgroup: two 16-row WMMA stripes
#define LDP     (L_DIM + 4) // LDS row pitch (pad 4 floats -> rotate banks)
#define TILES   8           // 16x16 col tiles per wave; 8 waves * 128 = 1024 cols

// ---------------------------------------------------------------------------
// Kernel 1: b2[m] = sum_l centroids[m][l]^2
// ---------------------------------------------------------------------------
__global__ __launch_bounds__(256) void som_b2_kernel(const float* __restrict__ cent,
                                                     float* __restrict__ b2) {
    int m = blockIdx.x * blockDim.x + threadIdx.x;
    if (m >= M_DIM) return;
    const float4* p = (const float4*)(cent + (size_t)m * L_DIM);
    float s = 0.f;
#pragma unroll 8
    for (int i = 0; i < L_DIM / 4; ++i) {
        float4 v = p[i];
        s += v.x * v.x + v.y * v.y + v.z * v.z + v.w * v.w;
    }
    b2[m] = s;
}

// ---------------------------------------------------------------------------
// Kernel 2: fused  wz = z*tw ;  G = wz @ cent^T (fp32 WMMA) ;
//           q = (1 + max(a2+b2-2G,0)/10)^-5.5 ;  row-normalize ; store
// 32 rows x 1024 cols per block; each wave: 2 row-stripes x 128 cols
// => each B-fragment load feeds 2 WMMAs.
// ---------------------------------------------------------------------------
__global__ __launch_bounds__(256) void som_main_kernel(const float* __restrict__ z,
                                                       const float* __restrict__ cent,
                                                       const float* __restrict__ tw,
                                                       const float* __restrict__ b2,
                                                       float* __restrict__ out) {
    __shared__ float lds_wz[ROWS * LDP];
    __shared__ float lds_a2[ROWS];
    __shared__ float lds_rs[ROWS];

    const int tid   = threadIdx.x;
    const int rbase = blockIdx.x * ROWS;

    if (tid < ROWS) { lds_a2[tid] = 0.f; lds_rs[tid] = 0.f; }
    __syncthreads();

    // ---- stage wz tile into LDS; accumulate a2 = sum(wz^2) per row --------
    {
        const int row = tid >> 3;            // 0..31 (8 threads per row)
        const int c0  = (tid & 7) * 32;      // 32 floats per thread
        const int rg  = rbase + row;
        const float w = tw[rg % T_SEQ];
        const float4* src = (const float4*)(z + (size_t)rg * L_DIM + c0);
        float* dst = &lds_wz[row * LDP + c0];
        float ss = 0.f;
#pragma unroll
        for (int i = 0; i < 8; ++i) {
            float4 v = src[i];
            v.x *= w; v.y *= w; v.z *= w; v.w *= w;
            ss += v.x * v.x + v.y * v.y + v.z * v.z + v.w * v.w;
            dst[i * 4 + 0] = v.x; dst[i * 4 + 1] = v.y;
            dst[i * 4 + 2] = v.z; dst[i * 4 + 3] = v.w;
        }
        atomicAdd(&lds_a2[row], ss);
    }
    __syncthreads();

    // ---- WMMA GEMM: (2 x 16 rows) x 128 cols per wave ---------------------
    const int wave = tid >> 5;               // 0..7
    const int lane = tid & 31;
    const int lm   = lane & 15;              // row (A) / col (B,C) within tile
    const int kb   = (lane >> 4) << 1;       // K sub-offset: 0 or 2
    const int n0w  = wave * (TILES * 16);    // this wave's column base

    v8f acc[2][TILES];
#pragma unroll
    for (int s = 0; s < 2; ++s)
#pragma unroll
        for (int t = 0; t < TILES; ++t)
            acc[s][t] = (v8f){0.f,0.f,0.f,0.f,0.f,0.f,0.f,0.f};

    const float* arow0 = &lds_wz[(lm     ) * LDP + kb];
    const float* arow1 = &lds_wz[(lm + 16) * LDP + kb];

    for (int k0 = 0; k0 < L_DIM; k0 += 4) {
        v2f a0 = *(const v2f*)(arow0 + k0);  // ds_load_b64, conflict-free (pad)
        v2f a1 = *(const v2f*)(arow1 + k0);
#pragma unroll
        for (int t = 0; t < TILES; ++t) {
            const float* bp = cent + (size_t)(n0w + t * 16 + lm) * L_DIM + k0 + kb;
            v2f b = *(const v2f*)bp;         // global_load_b64 (L2-resident), feeds 2 WMMAs
            acc[0][t] = __builtin_amdgcn_wmma_f32_16x16x4_f32(
                false, a0, false, b, (short)0, acc[0][t], false, false);
            acc[1][t] = __builtin_amdgcn_wmma_f32_16x16x4_f32(
                false, a1, false, b, (short)0, acc[1][t], false, false);
        }
    }

    // ---- epilogue: d2 -> q, per-row sums, normalize, store ----------------
    const int hi8 = (lane >> 4) << 3;        // row offset 0 or 8 (C layout)

    float rs[2][8];
#pragma unroll
    for (int s = 0; s < 2; ++s)
#pragma unroll
        for (int i = 0; i < 8; ++i) rs[s][i] = 0.f;

#pragma unroll
    for (int t = 0; t < TILES; ++t) {
        const float bc = b2[n0w + t * 16 + lm];
#pragma unroll
        for (int s = 0; s < 2; ++s) {
#pragma unroll
            for (int i = 0; i < 8; ++i) {
                float d2 = lds_a2[s * 16 + hi8 + i] + bc - 2.0f * acc[s][t][i];
                d2 = fmaxf(d2, 0.0f);
                // (1 + d2/10)^(-5.5) = exp2(-5.5 * log2(1 + d2*0.1))
                // v_log_f32 is log2, v_exp_f32 is exp2
                float q = __builtin_amdgcn_exp2f(-5.5f * __builtin_amdgcn_logf(1.0f + d2 * 0.1f));
                acc[s][t][i] = q;
                rs[s][i] += q;
            }
        }
    }

    // reduce row partial sums across each 16-lane half (columns of the tile)
#pragma unroll
    for (int m = 1; m <= 8; m <<= 1) {
#pragma unroll
        for (int s = 0; s < 2; ++s)
#pragma unroll
            for (int i = 0; i < 8; ++i) rs[s][i] += __shfl_xor(rs[s][i], m, 32);
    }
    if (lm == 0) {
#pragma unroll
        for (int s = 0; s < 2; ++s)
#pragma unroll
            for (int i = 0; i < 8; ++i) atomicAdd(&lds_rs[s * 16 + hi8 + i], rs[s][i]);
    }
    __syncthreads();

    float inv[2][8];
#pragma unroll
    for (int s = 0; s < 2; ++s)
#pragma unroll
        for (int i = 0; i < 8; ++i)
            inv[s][i] = 1.0f / (lds_rs[s * 16 + hi8 + i] + 1e-8f);

#pragma unroll
    for (int s = 0; s < 2; ++s) {
#pragma unroll
        for (int t = 0; t < TILES; ++t) {
#pragma unroll
            for (int i = 0; i < 8; ++i) {
                const size_t off =
                    (size_t)(rbase + s * 16 + hi8 + i) * M_DIM + (n0w + t * 16 + lm);
                out[off] = acc[s][t][i] * inv[s][i];
            }
        }
    }
}

// ---------------------------------------------------------------------------
extern "C" void kernel_launch(void* const* d_in, const int* in_sizes, int n_in,
                              void* d_out, int out_size, void* d_ws, size_t ws_size,
                              hipStream_t stream) {
    const float* z    = (const float*)d_in[0];  // (16, 3500, 256) f32
    const float* cent = (const float*)d_in[1];  // (1024, 256)     f32
    const float* tw   = (const float*)d_in[2];  // (1, 3500, 1)    f32
    float* out = (float*)d_out;                 // (16, 3500, 1024) f32
    float* b2  = (float*)d_ws;                  // 1024 floats scratch

    som_b2_kernel<<<dim3(M_DIM / 256), dim3(256), 0, stream>>>(cent, b2);

    const int nrows = 16 * T_SEQ;               // 56000
    som_main_kernel<<<dim3(nrows / ROWS), dim3(256), 0, stream>>>(z, cent, tw, b2, out);
}